// NeuronLevelModel_70677981823274
// MI455X (gfx1250) — compile-verified
//
#include <hip/hip_runtime.h>
#include <cmath>

// CDNA5 / gfx1250, wave32. fp32 WMMA (memory-bound; keep full precision) + TDM staging.

typedef __attribute__((ext_vector_type(2))) float v2f;
typedef __attribute__((ext_vector_type(8))) float v8f;
typedef unsigned int v4u __attribute__((ext_vector_type(4)));
typedef int v8i_t __attribute__((ext_vector_type(8)));
typedef int v4i_t __attribute__((ext_vector_type(4)));

#define NEURONS  2048
#define HIST     50
#define HIDDEN   64
#define BATCH    256
#define NKCHUNK  13          // ceil(50/4); last chunk: A garbage x B zeros = 0
#define W2STRIDE 160         // floats per k2-row: 128 data + 32 pad -> lane halves on disjoint banks

#if __has_builtin(__builtin_amdgcn_tensor_load_to_lds) && __has_builtin(__builtin_amdgcn_s_wait_tensorcnt)
#define USE_TDM 1
#else
#define USE_TDM 0
#endif

__device__ __forceinline__ float gelu_exact(float x) {
    return 0.5f * x * (1.0f + erff(x * 0.70710678118654752440f));
}

__global__ __launch_bounds__(256, 1)
void neuron_mlp_wmma_f32(const float* __restrict__ history,  // (BATCH, NEURONS, HIST)
                         const float* __restrict__ W1,       // (NEURONS, HIST, HIDDEN)
                         const float* __restrict__ b1,       // (NEURONS, HIDDEN)
                         const float* __restrict__ W2,       // (NEURONS, HIDDEN)
                         const float* __restrict__ b2,       // (NEURONS)
                         float* __restrict__ out)            // (BATCH, NEURONS)
{
    __shared__ float Alds[BATCH * HIST];          // packed 256x50  (51200 B) — TDM-friendly
    __shared__ float Wlds[26 * W2STRIDE];         // K-pair interleaved W1 (16640 B)
    __shared__ float b1lds[HIDDEN];
    __shared__ float w2lds[HIDDEN];

    const int n    = blockIdx.x;
    const int tid  = threadIdx.x;          // 0..255
    const int lane = tid & 31;
    const int wave = tid >> 5;             // 0..7
    const int half = lane >> 4;            // K-pair select (ISA 16x4 f32 A/B layout)
    const int l16  = lane & 15;

#if USE_TDM
    // ---- issue TDM: 2D tile (rows=256 batch, cols=50) of history[:, n, :] -> packed Alds ----
    if (tid == 0) {
        unsigned long long gaddr =
            (unsigned long long)(uintptr_t)history + (unsigned long long)n * (HIST * 4);
        __attribute__((address_space(3))) float* a3 =
            (__attribute__((address_space(3))) float*)&Alds[0];
        unsigned int lds_off = (unsigned int)(uintptr_t)a3;

        // D# group0: count=1 | lds_addr | global_addr[56:0] | type=2
        v4u g0 = { 1u, lds_off, (unsigned int)gaddr,
                   (unsigned int)((gaddr >> 32) & 0x01FFFFFFu) | (2u << 30) };
        // D# group1: data_size=4B; tensor_dim0=50; tensor_dim1=256; tile_dim0=50;
        //            tile_dim1=256; tile_dim2=0; tensor_dim0_stride=NEURONS*HIST
        v8i_t g1 = { (int)0x00020000,
                     (int)(HIST  << 16),
                     (int)(BATCH << 16),
                     (int)(HIST  << 16),
                     (int)BATCH,
                     (int)(NEURONS * HIST),
                     0, 0 };
        v4i_t gz = {};
#if __has_include(<hip/amd_detail/amd_gfx1250_TDM.h>)
        v8i_t gz8 = {};
        __builtin_amdgcn_tensor_load_to_lds(g0, g1, gz, gz, gz8, 0);
#else
        __builtin_amdgcn_tensor_load_to_lds(g0, g1, gz, gz, 0);
#endif
    }
#else
    // ---- fallback: per-thread copy of history row `tid` (50 floats, 8B aligned) ----
    {
        const float2* src = reinterpret_cast<const float2*>(
            history + ((size_t)tid * NEURONS + (size_t)n) * HIST);
        float2* dst = reinterpret_cast<float2*>(&Alds[tid * HIST]);
        #pragma unroll
        for (int i = 0; i < HIST / 2; ++i) dst[i] = src[i];
    }
#endif

    // ---- stage W1[n] (50x64, contiguous, 16B aligned) K-pair interleaved:
    //      Wlds[(K/2)*W2STRIDE + col*2 + (K&1)]  -> B frag = one ds_load_b64 ----
    {
        const float4* src = reinterpret_cast<const float4*>(W1 + (size_t)n * (HIST * HIDDEN));
        for (int i = tid; i < (HIST * HIDDEN) / 4; i += 256) {
            float4 v = src[i];
            int t  = i >> 4;             // K row 0..49
            int c  = (i & 15) << 2;      // col
            float* d = &Wlds[(t >> 1) * W2STRIDE + (c << 1) + (t & 1)];
            d[0] = v.x; d[2] = v.y; d[4] = v.z; d[6] = v.w;
        }
        if (tid < 128) Wlds[25 * W2STRIDE + tid] = 0.0f;   // zero K=50,51 slots
        if (tid < HIDDEN) {
            b1lds[tid] = b1[(size_t)n * HIDDEN + tid];
            w2lds[tid] = W2[(size_t)n * HIDDEN + tid];
        }
    }

#if USE_TDM
    if (tid == 0) __builtin_amdgcn_s_wait_tensorcnt(0);
#endif
    __syncthreads();

    const float bias2 = b2[n];

    // hoisted epilogue constants (depend only on l16)
    float bb[4], ww[4];
    #pragma unroll
    for (int t = 0; t < 4; ++t) {
        bb[t] = b1lds[t * 16 + l16];
        ww[t] = w2lds[t * 16 + l16];
    }

    // Each wave: 2 M-tiles (16 batch rows) x all 4 N-tiles (16 hidden cols).
    #pragma unroll
    for (int mi = 0; mi < 2; ++mi) {
        const int mt = wave * 2 + mi;
        const float* Arow = &Alds[(mt * 16 + l16) * HIST + half * 2];
        const float* Brow = &Wlds[half * W2STRIDE + (l16 << 1)];

        v8f acc0 = {}, acc1 = {}, acc2 = {}, acc3 = {};

        #pragma unroll
        for (int k = 0; k < NKCHUNK; ++k) {
            // A frag: A[M = mt*16+l16][K = 4k+2*half .. +1]  (one b64)
            v2f a = *reinterpret_cast<const v2f*>(Arow + 4 * k);
            // B frags: {B[K0][c], B[K0+1][c]} adjacent in LDS (one b64 each)
            const float* Bk = Brow + 2 * k * W2STRIDE;
            v2f bf0 = *reinterpret_cast<const v2f*>(Bk);
            v2f bf1 = *reinterpret_cast<const v2f*>(Bk + 32);
            v2f bf2 = *reinterpret_cast<const v2f*>(Bk + 64);
            v2f bf3 = *reinterpret_cast<const v2f*>(Bk + 96);
            acc0 = __builtin_amdgcn_wmma_f32_16x16x4_f32(false, a, false, bf0, (short)0, acc0, false, false);
            acc1 = __builtin_amdgcn_wmma_f32_16x16x4_f32(false, a, false, bf1, (short)0, acc1, false, false);
            acc2 = __builtin_amdgcn_wmma_f32_16x16x4_f32(false, a, false, bf2, (short)0, acc2, false, false);
            acc3 = __builtin_amdgcn_wmma_f32_16x16x4_f32(false, a, false, bf3, (short)0, acc3, false, false);
        }

        // Epilogue: h = gelu(acc + b1); s = h*W2; reduce across 64 hidden cols.
        float rowsum[8];
        #pragma unroll
        for (int r = 0; r < 8; ++r) {
            float s = gelu_exact(acc0[r] + bb[0]) * ww[0];
            s      += gelu_exact(acc1[r] + bb[1]) * ww[1];
            s      += gelu_exact(acc2[r] + bb[2]) * ww[2];
            s      += gelu_exact(acc3[r] + bb[3]) * ww[3];
            // butterfly within each 16-lane half (two independent batch rows)
            s += __shfl_xor(s, 8, 32);
            s += __shfl_xor(s, 4, 32);
            s += __shfl_xor(s, 2, 32);
            s += __shfl_xor(s, 1, 32);
            rowsum[r] = s;
        }

        if (l16 == 0) {
            const int rowbase = mt * 16 + half * 8;
            #pragma unroll
            for (int r = 0; r < 8; ++r)
                out[(size_t)(rowbase + r) * NEURONS + n] = rowsum[r] + bias2;
        }
    }
}

extern "C" void kernel_launch(void* const* d_in, const int* in_sizes, int n_in,
                              void* d_out, int out_size, void* d_ws, size_t ws_size,
                              hipStream_t stream) {
    (void)in_sizes; (void)n_in; (void)d_ws; (void)ws_size; (void)out_size;
    const float* history = (const float*)d_in[0];
    const float* W1      = (const float*)d_in[1];
    const float* b1      = (const float*)d_in[2];
    const float* W2      = (const float*)d_in[3];
    const float* b2      = (const float*)d_in[4];
    float* out           = (float*)d_out;

    neuron_mlp_wmma_f32<<<dim3(NEURONS), dim3(256), 0, stream>>>(
        history, W1, b1, W2, b2, out);
}